// Net_32762010534229
// MI455X (gfx1250) — compile-verified
//
#include <hip/hip_runtime.h>

// SGConv reformulated: out = P^3 (X @ w) + b  where P = D^-1/2 (A+I) D^-1/2.
// Project-first collapses per-hop payload 64x; all hop state is L2-resident.

#define N_NODES 50000
#define N_EDGES 800000
#define N_FEATS 64

typedef __attribute__((ext_vector_type(2))) float v2f;
typedef __attribute__((ext_vector_type(8))) float v8f;

// ---------------- degree / normalization ----------------

__global__ void init_deg_kernel(float* __restrict__ deg, int n) {
    int i = blockIdx.x * blockDim.x + threadIdx.x;
    if (i < n) deg[i] = 1.0f;  // self-loop contributes 1 to every destination
}

__global__ void deg_edges_kernel(const int* __restrict__ ei, float* __restrict__ deg, int E) {
    int e = blockIdx.x * blockDim.x + threadIdx.x;
    if (e < E) atomicAdd(&deg[ei[E + e]], 1.0f);  // col = edge_index[1][e]
}

__global__ void rsqrt_kernel(float* __restrict__ d, int n) {
    int i = blockIdx.x * blockDim.x + threadIdx.x;
    if (i < n) d[i] = rsqrtf(d[i]);  // deg >= 1 always (self-loops), no zero guard needed
}

// ---------------- y = X @ w via V_WMMA_F32_16X16X4_F32 ----------------
// One wave per 16-row tile (grid = N/16 blocks of 32 threads; N % 16 == 0).
// B has w broadcast into every column, so all 16 output columns equal X@w.
// C layout: lane 0 VGPR j -> (M=j, N=0); lane 16 VGPR j -> (M=8+j, N=0).

__global__ void gemv_wmma_kernel(const float* __restrict__ x,
                                 const float* __restrict__ w,
                                 float* __restrict__ y) {
    int base = blockIdx.x * 16;
    int lane = threadIdx.x & 31;
    int m    = lane & 15;
    int koff = (lane < 16) ? 0 : 2;
    const float* xrow = x + (size_t)(base + m) * N_FEATS;

#if __has_builtin(__builtin_amdgcn_wmma_f32_16x16x4_f32)
    v8f c = {};
#pragma unroll
    for (int k0 = 0; k0 < N_FEATS; k0 += 4) {
        // A: 16x4 f32 tile. lane m = row, VGPR0/1 = K koff / koff+1.
        v2f a = *(const v2f*)(xrow + k0 + koff);
        // B: 4x16 f32, every column = w (broadcast), same K striping as A.
        v2f bb;
        bb.x = w[k0 + koff];
        bb.y = w[k0 + koff + 1];
        c = __builtin_amdgcn_wmma_f32_16x16x4_f32(
            /*neg_a=*/false, a, /*neg_b=*/false, bb,
            /*c_mod=*/(short)0, c, /*reuse_a=*/false, /*reuse_b=*/false);
    }
    if (m == 0) {  // lanes 0 and 16 hold column N=0 of the result
        float* yo = y + base + (lane >> 4) * 8;
#pragma unroll
        for (int j = 0; j < 8; ++j) yo[j] = c[j];
    }
#else
    // Scalar fallback (should not be taken on gfx1250).
    if (lane < 16) {
        float acc = 0.0f;
#pragma unroll
        for (int k = 0; k < N_FEATS; ++k) acc += xrow[k] * w[k];
        y[base + m] = acc;
    }
#endif
}

// ---------------- one propagation hop ----------------

__global__ void hop_init_kernel(const float* __restrict__ dinv,
                                const float* __restrict__ ysrc,
                                float* __restrict__ ydst,
                                const float* __restrict__ bias, int n) {
    int i = blockIdx.x * blockDim.x + threadIdx.x;
    if (i < n) {
        float di = dinv[i];
        float v  = di * di * ysrc[i];  // self-loop term: norm = dinv[i]*dinv[i]
        if (bias) v += bias[0];        // fold +b into the final hop (commutes with atomics)
        ydst[i] = v;
    }
}

__global__ void hop_edges_kernel(const int* __restrict__ ei,
                                 const float* __restrict__ dinv,
                                 const float* __restrict__ ysrc,
                                 float* __restrict__ ydst, int E) {
    int e = blockIdx.x * blockDim.x + threadIdx.x;
    if (e < E) {
        int r = ei[e];        // source (row)
        int c = ei[E + e];    // destination (col)
        atomicAdd(&ydst[c], dinv[r] * dinv[c] * ysrc[r]);
    }
}

// ---------------- launch ----------------

extern "C" void kernel_launch(void* const* d_in, const int* in_sizes, int n_in,
                              void* d_out, int out_size, void* d_ws, size_t ws_size,
                              hipStream_t stream) {
    (void)in_sizes; (void)n_in; (void)out_size; (void)ws_size;

    const float* x  = (const float*)d_in[0];  // [N, 64]
    const int*   ei = (const int*)  d_in[1];  // [2, E] flat
    const float* W  = (const float*)d_in[2];  // [1, 64]
    const float* b  = (const float*)d_in[3];  // [1]
    float*       out = (float*)d_out;         // [N]

    const int N = N_NODES;
    const int E = N_EDGES;

    // Workspace: dinv | yA | yB  (16-float aligned stride), ~600 KB total.
    const int S = 50048;
    float* f    = (float*)d_ws;
    float* dinv = f;
    float* yA   = f + S;
    float* yB   = f + 2 * S;

    dim3 blk(256);
    dim3 gn((N + 255) / 256);
    dim3 ge((E + 255) / 256);

    // 1) degrees -> dinv (in place)
    init_deg_kernel<<<gn, blk, 0, stream>>>(dinv, N);
    deg_edges_kernel<<<ge, blk, 0, stream>>>(ei, dinv, E);
    rsqrt_kernel<<<gn, blk, 0, stream>>>(dinv, N);

    // 2) y0 = X @ w   (WMMA f32 16x16x4; one wave per 16 rows)
    gemv_wmma_kernel<<<N / 16, 32, 0, stream>>>(x, W, yA);

    // 3) three hops of y <- P y (ping-pong; last hop writes d_out with +b)
    hop_init_kernel <<<gn, blk, 0, stream>>>(dinv, yA, yB, nullptr, N);
    hop_edges_kernel<<<ge, blk, 0, stream>>>(ei, dinv, yA, yB, E);

    hop_init_kernel <<<gn, blk, 0, stream>>>(dinv, yB, yA, nullptr, N);
    hop_edges_kernel<<<ge, blk, 0, stream>>>(ei, dinv, yB, yA, E);

    hop_init_kernel <<<gn, blk, 0, stream>>>(dinv, yA, out, b, N);
    hop_edges_kernel<<<ge, blk, 0, stream>>>(ei, dinv, yA, out, E);
}